// GCN_43868795961418
// MI455X (gfx1250) — compile-verified
//
#include <hip/hip_runtime.h>

typedef __attribute__((ext_vector_type(2))) float v2f;
typedef __attribute__((ext_vector_type(8))) float v8f;

#define BN_EPS 1e-5f

// ----------------------------------------------------------------------------
// Graph-norm kernels
// ----------------------------------------------------------------------------
__global__ void k_fill(float* __restrict__ p, float v, int n) {
  int i = blockIdx.x * blockDim.x + threadIdx.x;
  if (i < n) p[i] = v;
}

__global__ void k_deg(const int* __restrict__ dst, const float* __restrict__ attr,
                      float* __restrict__ deg, int e) {
  int i = blockIdx.x * blockDim.x + threadIdx.x;
  if (i < e) atomicAdd(&deg[dst[i]], attr[i]);
}

// deg (stored in dinv) -> dinv = rsqrt(deg+1), selfn = dinv*dinv
__global__ void k_node_norm(float* __restrict__ dinv, float* __restrict__ selfn, int n) {
  int i = blockIdx.x * blockDim.x + threadIdx.x;
  if (i < n) {
    float di = rsqrtf(dinv[i] + 1.0f);
    dinv[i] = di;
    selfn[i] = di * di;
  }
}

__global__ void k_edge_norm(const int* __restrict__ src, const int* __restrict__ dst,
                            const float* __restrict__ attr, const float* __restrict__ dinv,
                            float* __restrict__ wn, int e) {
  int i = blockIdx.x * blockDim.x + threadIdx.x;
  if (i < e) wn[i] = dinv[src[i]] * attr[i] * dinv[dst[i]];
}

// ----------------------------------------------------------------------------
// Async global->LDS copy of one 16-byte chunk (gfx1250 GLOBAL_LOAD_ASYNC_TO_LDS,
// tracked by ASYNCcnt). lds_off is the wave-relative LDS byte address (low 32
// bits of the generic pointer), gptr a 16B-aligned global address.
// ----------------------------------------------------------------------------
__device__ __forceinline__ void async_copy_b128(unsigned int lds_off, const float* gptr) {
  asm volatile("global_load_async_to_lds_b128 %0, %1, off"
               :: "v"(lds_off), "v"(gptr)
               : "memory");
}

__device__ __forceinline__ void async_wait_all() {
  asm volatile("s_wait_asynccnt 0x0" ::: "memory");
}

// ----------------------------------------------------------------------------
// WMMA GEMM: per block, 16 rows of X (N x 128) times W (128 x c_out).
// CT = number of 16-wide column tiles (c_out padded to CT*16). blockDim = CT*32.
// Wave w computes the 16x16 tile at columns [w*16, w*16+16).
// Writes h = X@W and out = selfn[m]*h + bias[n].
// V_WMMA_F32_16X16X4_F32, 32 K-steps over K=128. X/W staged into LDS via
// async-to-LDS B128 copies. Padded W columns (>= c_out) are left uninitialized:
// garbage there only affects output columns that are never stored.
// ----------------------------------------------------------------------------
template <int CT>
__global__ __launch_bounds__(CT * 32) void k_gcn_gemm(
    const float* __restrict__ X, const float* __restrict__ W,
    const float* __restrict__ bias, const float* __restrict__ selfn,
    float* __restrict__ h, float* __restrict__ out, int c_out) {
  constexpr int CP = CT * 16;
  constexpr int WSTR = CP + 4;        // +4 floats: 16B-aligned rows, banks offset by 4/row
  __shared__ __align__(16) float Ws[128][WSTR];
  __shared__ __align__(16) float Xs[16][132];   // 528B rows: 16B aligned, conflict-free

  const int tid = threadIdx.x;
  const int nthr = CT * 32;
  const int mbase = blockIdx.x * 16;

  const unsigned int ws_base = (unsigned int)(size_t)(&Ws[0][0]);
  const unsigned int xs_base = (unsigned int)(size_t)(&Xs[0][0]);

  // Stage W: 128 rows x (c_out/4) 16B chunks, raw async copy (no zero pad needed).
  const int cpr = c_out >> 2;  // 16B chunks per W row (32 for 128, 10 for 40)
  for (int i = tid; i < 128 * cpr; i += nthr) {
    const int k = i / cpr, cc = i - k * cpr;
    async_copy_b128(ws_base + (unsigned int)(k * WSTR + cc * 4) * 4u,
                    W + (size_t)k * c_out + cc * 4);
  }
  // Stage 16x128 X tile: 16 rows x 32 chunks.
  for (int i = tid; i < 16 * 32; i += nthr) {
    const int r = i >> 5, cc = i & 31;
    async_copy_b128(xs_base + (unsigned int)(r * 132 + cc * 4) * 4u,
                    X + (size_t)(mbase + r) * 128 + cc * 4);
  }
  async_wait_all();   // this wave's async LDS writes complete
  __syncthreads();    // all waves' writes visible

  const int wave = tid >> 5;
  const int lane = tid & 31;
  const int l15 = lane & 15;
  const int half = lane >> 4;          // lanes 16-31 carry the upper K pair / M+8 rows
  const int nloc = wave * 16 + l15;

  v8f acc = {};
#pragma unroll
  for (int kt = 0; kt < 32; ++kt) {
    const int kb = kt * 4 + half * 2;  // A/B 16x4 f32 layout: lo lanes K={0,1}, hi K={2,3}
    v2f a, b;
    a[0] = Xs[l15][kb];
    a[1] = Xs[l15][kb + 1];
    b[0] = Ws[kb][nloc];
    b[1] = Ws[kb + 1][nloc];
    acc = __builtin_amdgcn_wmma_f32_16x16x4_f32(false, a, false, b, (short)0, acc,
                                                false, false);
  }

  // C/D layout: VGPR r -> M = r (lanes 0-15) or r+8 (lanes 16-31); N = lane&15
  const int n = wave * 16 + l15;
  if (n < c_out) {
    const float bv = bias[n];
#pragma unroll
    for (int r = 0; r < 8; ++r) {
      const int m = mbase + r + 8 * half;
      const float hv = acc[r];
      const size_t o = (size_t)m * c_out + n;
      h[o] = hv;
      out[o] = selfn[m] * hv + bv;
    }
  }
}

// ----------------------------------------------------------------------------
// Edge scatter: one wave per edge. out[dst] += h[src] * w_norm (atomic f32).
// ----------------------------------------------------------------------------
__global__ void k_scatter128(const float* __restrict__ h, const float* __restrict__ wn,
                             const int* __restrict__ src, const int* __restrict__ dst,
                             float* __restrict__ out, int e) {
  const int gid = blockIdx.x * blockDim.x + threadIdx.x;
  const int edge = gid >> 5;
  const int lane = threadIdx.x & 31;
  if (edge >= e) return;
  const int s = src[edge];
  const int d = dst[edge];
  const float w = wn[edge];
  const float4 v = ((const float4*)(h + (size_t)s * 128))[lane];
  float* op = out + (size_t)d * 128 + lane * 4;
  atomicAdd(op + 0, v.x * w);
  atomicAdd(op + 1, v.y * w);
  atomicAdd(op + 2, v.z * w);
  atomicAdd(op + 3, v.w * w);
}

__global__ void k_scatter40(const float* __restrict__ h, const float* __restrict__ wn,
                            const int* __restrict__ src, const int* __restrict__ dst,
                            float* __restrict__ out, int e) {
  const int gid = blockIdx.x * blockDim.x + threadIdx.x;
  const int edge = gid >> 5;
  const int lane = threadIdx.x & 31;
  if (edge >= e) return;
  const int s = src[edge];
  const int d = dst[edge];
  const float w = wn[edge];
  const float* hp = h + (size_t)s * 40;
  float* op = out + (size_t)d * 40;
  atomicAdd(op + lane, hp[lane] * w);
  if (lane < 8) atomicAdd(op + 32 + lane, hp[32 + lane] * w);
}

// ----------------------------------------------------------------------------
// BatchNorm (training-mode batch stats over node dim) + ReLU
// ----------------------------------------------------------------------------
__global__ void k_bn_stats(const float* __restrict__ x, float* __restrict__ stats,
                           int nrows) {
  const int c = threadIdx.x;  // 128 channels, 128 threads
  float s = 0.0f, s2 = 0.0f;
  for (int r = blockIdx.x; r < nrows; r += gridDim.x) {
    const float v = x[(size_t)r * 128 + c];
    s += v;
    s2 += v * v;
  }
  atomicAdd(&stats[c], s);
  atomicAdd(&stats[128 + c], s2);
}

__global__ void k_bn_apply(float* __restrict__ x, const float* __restrict__ stats,
                           const float* __restrict__ g, const float* __restrict__ bt,
                           int total, float invn) {
  const int i = blockIdx.x * blockDim.x + threadIdx.x;
  if (i >= total) return;
  const int c = i & 127;
  const float mu = stats[c] * invn;
  const float var = stats[128 + c] * invn - mu * mu;
  const float y = g[c] * (x[i] - mu) * rsqrtf(var + BN_EPS) + bt[c];
  x[i] = fmaxf(y, 0.0f);
}

// ----------------------------------------------------------------------------
// Host launcher
// ----------------------------------------------------------------------------
extern "C" void kernel_launch(void* const* d_in, const int* in_sizes, int n_in,
                              void* d_out, int out_size, void* d_ws, size_t ws_size,
                              hipStream_t stream) {
  const float* x    = (const float*)d_in[0];
  const float* attr = (const float*)d_in[1];
  const float* W1   = (const float*)d_in[2];
  const float* b1   = (const float*)d_in[3];
  const float* g1   = (const float*)d_in[4];
  const float* bt1  = (const float*)d_in[5];
  const float* W2   = (const float*)d_in[6];
  const float* b2   = (const float*)d_in[7];
  const float* g2   = (const float*)d_in[8];
  const float* bt2  = (const float*)d_in[9];
  const float* W3   = (const float*)d_in[10];
  const float* b3   = (const float*)d_in[11];
  const int*   ei   = (const int*)d_in[12];

  const int N = in_sizes[0] / 128;
  const int E = in_sizes[1];
  const int OUTC = 40;
  const int G = 2;
  const int* src = ei;
  const int* dst = ei + E;

  // Workspace carve-up (floats)
  float* ws = (float*)d_ws;
  float* wn    = ws;                         // E
  float* dinv  = wn + E;                     // N (deg, then dinv in place)
  float* selfn = dinv + N;                   // N
  float* stats = selfn + N;                  // 256 (sum[128], sumsq[128])
  float* bufA  = stats + 256;                // N*128  (h scratch)
  float* bufB  = bufA + (size_t)N * 128;     // N*128  (conv1 out / conv2 in)
  float* bufC  = bufB + (size_t)N * 128;     // N*128  (conv2 out / conv3 in)
  (void)ws_size; (void)n_in; (void)out_size;

  const int TPB = 256;
  const int nb_N = (N + TPB - 1) / TPB;
  const int nb_E = (E + TPB - 1) / TPB;
  const int nb_edgewave = (E * 32 + TPB - 1) / TPB;  // one wave per edge
  const int nb_elem = (N * 128 + TPB - 1) / TPB;
  const int nb_gemm = N / 16;  // N = 100000, divisible by 16
  const float invn = 1.0f / (float)N;

  // --- gcn_norm (once) ---
  k_fill<<<nb_N, TPB, 0, stream>>>(dinv, 0.0f, N);
  k_deg<<<nb_E, TPB, 0, stream>>>(dst, attr, dinv, E);
  k_node_norm<<<nb_N, TPB, 0, stream>>>(dinv, selfn, N);
  k_edge_norm<<<nb_E, TPB, 0, stream>>>(src, dst, attr, dinv, wn, E);

  for (int gi = 0; gi < G; ++gi) {
    const float* W1g = W1 + (size_t)gi * 128 * 128;
    const float* W2g = W2 + (size_t)gi * 128 * 128;
    const float* W3g = W3 + (size_t)gi * 128 * OUTC;
    const float* b1g = b1 + gi * 128;
    const float* b2g = b2 + gi * 128;
    const float* b3g = b3 + gi * OUTC;
    const float* g1g = g1 + gi * 128;
    const float* g2g = g2 + gi * 128;
    const float* bt1g = bt1 + gi * 128;
    const float* bt2g = bt2 + gi * 128;
    float* outg = (float*)d_out + (size_t)gi * N * OUTC;

    // conv1: h=x@W1 -> bufA ; out=selfn*h+b -> bufB ; scatter into bufB
    k_gcn_gemm<8><<<nb_gemm, 256, 0, stream>>>(x, W1g, b1g, selfn, bufA, bufB, 128);
    k_scatter128<<<nb_edgewave, TPB, 0, stream>>>(bufA, wn, src, dst, bufB, E);
    k_fill<<<1, 256, 0, stream>>>(stats, 0.0f, 256);
    k_bn_stats<<<1024, 128, 0, stream>>>(bufB, stats, N);
    k_bn_apply<<<nb_elem, TPB, 0, stream>>>(bufB, stats, g1g, bt1g, N * 128, invn);

    // conv2: input bufB -> h bufA, out bufC
    k_gcn_gemm<8><<<nb_gemm, 256, 0, stream>>>(bufB, W2g, b2g, selfn, bufA, bufC, 128);
    k_scatter128<<<nb_edgewave, TPB, 0, stream>>>(bufA, wn, src, dst, bufC, E);
    k_fill<<<1, 256, 0, stream>>>(stats, 0.0f, 256);
    k_bn_stats<<<1024, 128, 0, stream>>>(bufC, stats, N);
    k_bn_apply<<<nb_elem, TPB, 0, stream>>>(bufC, stats, g2g, bt2g, N * 128, invn);

    // conv3: input bufC -> h bufA (stride 40), out = final output slice
    k_gcn_gemm<3><<<nb_gemm, 96, 0, stream>>>(bufC, W3g, b3g, selfn, bufA, outg, OUTC);
    k_scatter40<<<nb_edgewave, TPB, 0, stream>>>(bufA, wn, src, dst, outg, E);
  }
}